// ContrastiveTripletLoss_35871566856560
// MI455X (gfx1250) — compile-verified
//
#include <hip/hip_runtime.h>
#include <hip/hip_bf16.h>

typedef __attribute__((ext_vector_type(2))) float v2f;
typedef __attribute__((ext_vector_type(8))) float v8f;

#define N_ 4
#define C_ 24
#define E_ 16
#define H_ 768
#define W_ 768
#define P_ (H_ * W_)          /* 589824 pixels per sample */
#define TILES_ (P_ / 32)      /* 18432 tiles of 32 pixels */

/* workspace layout in floats */
#define WS_SUMS   0           /* N*C*E = 1536 */
#define WS_COUNTS 1536        /* N*C   =   96 */
#define WS_MEANS  1632        /* N*C*E = 1536 */
#define WS_VAR    3168        /* N*C   =   96 */
#define WS_REG    3264        /* N     =    4 */
#define WS_TOTAL  3268

__global__ void zero_k(float* __restrict__ ws, float* __restrict__ out) {
  int t = blockIdx.x * blockDim.x + threadIdx.x;
  for (int i = t; i < WS_TOTAL; i += blockDim.x * gridDim.x) ws[i] = 0.f;
  if (t == 0) out[0] = 0.f;
}

/* Pass 1: per-cluster sums via one-hot WMMA + counts histogram.
 * grid = (72, 4), block = 256 (8 waves). Each wave owns a private LDS tile of
 * 32 pixels x 16 channels and accumulates a 16x16 (clusters x E) C-fragment
 * per cluster-halftile with V_WMMA_F32_16X16X4_F32. */
__global__ __launch_bounds__(256) void segsum_k(const float* __restrict__ x,
                                                const int* __restrict__ target,
                                                float* __restrict__ sums,
                                                float* __restrict__ counts) {
  __shared__ float xs_all[8 * 32 * 17];   /* pad 17 -> conflict-free */
  __shared__ int   ss_all[8 * 32];
  __shared__ int   cnt_lds[C_];

  const int tid  = threadIdx.x;
  const int lane = tid & 31;
  const int wv   = tid >> 5;
  const int n    = blockIdx.y;

  if (tid < C_) cnt_lds[tid] = 0;
  __syncthreads();

  float* xs = xs_all + wv * (32 * 17);
  int*   ss = ss_all + wv * 32;

  const int  ch = lane & 15;
  const bool hi = lane >= 16;
  const int  waves = gridDim.x * 8;
  const int  wgid  = blockIdx.x * 8 + wv;

  v8f acc0 = {0.f, 0.f, 0.f, 0.f, 0.f, 0.f, 0.f, 0.f};
  v8f acc1 = {0.f, 0.f, 0.f, 0.f, 0.f, 0.f, 0.f, 0.f};

  const size_t xbase = (size_t)n * E_ * P_;
  const int*   tgt   = target + (size_t)n * P_;

  for (int tile = wgid; tile < TILES_; tile += waves) {
    const int p = tile * 32 + lane;
    const int s = tgt[p];
    ss[lane] = s;
    atomicAdd(&cnt_lds[s], 1);
#pragma unroll
    for (int e = 0; e < E_; ++e)            /* coalesced per-channel streams */
      xs[lane * 17 + e] = x[xbase + (size_t)e * P_ + p];

    /* 8 K-chunks of 4 pixels; two cluster halftiles (0-15, 16-31) */
#pragma unroll
    for (int k4 = 0; k4 < 8; ++k4) {
      const int kb = k4 * 4 + (hi ? 2 : 0);
      v2f B;
      B.x = xs[kb * 17 + ch];
      B.y = xs[(kb + 1) * 17 + ch];
      const int s0 = ss[kb];
      const int s1 = ss[kb + 1];
      v2f A0, A1;
      A0.x = (s0 == ch)      ? 1.f : 0.f;
      A0.y = (s1 == ch)      ? 1.f : 0.f;
      A1.x = (s0 == ch + 16) ? 1.f : 0.f;
      A1.y = (s1 == ch + 16) ? 1.f : 0.f;
      acc0 = __builtin_amdgcn_wmma_f32_16x16x4_f32(false, A0, false, B,
                                                   (short)0, acc0, false, false);
      acc1 = __builtin_amdgcn_wmma_f32_16x16x4_f32(false, A1, false, B,
                                                   (short)0, acc1, false, false);
    }
  }

  /* flush C fragments: VGPR i holds M=i (lanes 0-15) / M=i+8 (lanes 16-31), N=ch */
  const int mb = hi ? 8 : 0;
  float* sbase = sums + (size_t)n * C_ * E_;
#pragma unroll
  for (int i = 0; i < 8; ++i) {
    atomicAdd(&sbase[(mb + i) * E_ + ch], acc0[i]);
    const int m1 = 16 + mb + i;
    if (m1 < C_) atomicAdd(&sbase[m1 * E_ + ch], acc1[i]);
  }
  __syncthreads();
  if (tid < C_) atomicAdd(&counts[n * C_ + tid], (float)cnt_lds[tid]);
}

/* means = sums / counts ; regularizer (||mean|| - 1)^2 / C per sample */
__global__ void means_k(const float* __restrict__ sums,
                        const float* __restrict__ counts,
                        float* __restrict__ means,
                        float* __restrict__ reg) {
  int c = blockIdx.x * blockDim.x + threadIdx.x;
  if (c >= N_ * C_) return;
  const float inv = 1.f / counts[c];
  float nrm2 = 0.f;
#pragma unroll
  for (int e = 0; e < E_; ++e) {
    float m = sums[c * E_ + e] * inv;
    means[c * E_ + e] = m;
    nrm2 += m * m;
  }
  float r = sqrtf(nrm2) - 1.f;
  atomicAdd(&reg[c / C_], r * r * (1.f / C_));
}

/* Pass 2: hinged variance term. grid = (144, 4), block = 256. */
__global__ __launch_bounds__(256) void var_k(const float* __restrict__ x,
                                             const int* __restrict__ target,
                                             const float* __restrict__ means,
                                             float* __restrict__ varsum) {
  __shared__ float cm[C_ * E_];
  __shared__ float vloc[C_];
  const int tid = threadIdx.x;
  const int n   = blockIdx.y;
  for (int i = tid; i < C_ * E_; i += blockDim.x) cm[i] = means[n * C_ * E_ + i];
  if (tid < C_) vloc[tid] = 0.f;
  __syncthreads();

  const size_t xbase = (size_t)n * E_ * P_;
  const int*   tgt   = target + (size_t)n * P_;
  const int stride = gridDim.x * blockDim.x;
  for (int p = blockIdx.x * blockDim.x + tid; p < P_; p += stride) {
    const int s = tgt[p];
    float d2 = 0.f;
#pragma unroll
    for (int e = 0; e < E_; ++e) {
      float diff = x[xbase + (size_t)e * P_ + p] - cm[s * E_ + e];
      d2 += diff * diff;
    }
    float h = fmaxf(sqrtf(d2) - 0.5f, 0.f);
    atomicAdd(&vloc[s], h * h);
  }
  __syncthreads();
  if (tid < C_) atomicAdd(&varsum[n * C_ + tid], vloc[tid]);
}

/* Triplet term + final scalar. grid = 4 (one block per sample), block = 256. */
__global__ __launch_bounds__(256) void final_k(const float* __restrict__ means,
                                               const float* __restrict__ varsum,
                                               const float* __restrict__ counts,
                                               const float* __restrict__ reg,
                                               const int* __restrict__ e_attr,
                                               const int* __restrict__ e_rep,
                                               float* __restrict__ out) {
  __shared__ float cm[C_ * E_];
  __shared__ float dA[200], dR[200];
  __shared__ int a0s[200], a1s[200], r0s[200], r1s[200];
  __shared__ float sAcc, cAcc;
  const int tid = threadIdx.x;
  const int n   = blockIdx.x;

  for (int i = tid; i < C_ * E_; i += blockDim.x) cm[i] = means[n * C_ * E_ + i];
  if (tid == 0) { sAcc = 0.f; cAcc = 0.f; }
  __syncthreads();

  if (tid < 200) {
    int a0 = e_attr[n * 400 + tid];
    int a1 = e_attr[n * 400 + 200 + tid];
    float d = 0.f;
#pragma unroll
    for (int e = 0; e < E_; ++e) {
      float t = cm[a0 * E_ + e] - cm[a1 * E_ + e] + 1e-6f;
      d += t * t;
    }
    dA[tid] = d; a0s[tid] = a0; a1s[tid] = a1;
    int r0 = e_rep[n * 400 + tid];
    int r1 = e_rep[n * 400 + 200 + tid];
    d = 0.f;
#pragma unroll
    for (int e = 0; e < E_; ++e) {
      float t = cm[r0 * E_ + e] - cm[r1 * E_ + e] + 1e-6f;
      d += t * t;
    }
    dR[tid] = d; r0s[tid] = r0; r1s[tid] = r1;
  }
  __syncthreads();

  float lsum = 0.f;
  int   lcnt = 0;
  for (int idx = tid; idx < 200 * 200; idx += blockDim.x) {
    int a = idx / 200, r = idx % 200;
    int m = (a0s[a] == r0s[r]) + (a0s[a] == r1s[r]) +
            (a1s[a] == r0s[r]) + (a1s[a] == r1s[r]);
    if (m == 1) {
      float t = fmaxf(0.5f * (dA[a] - dR[r]) + 0.01f, 0.f);
      if (t > 0.f) { lsum += t; lcnt += 1; }
    }
  }
  atomicAdd(&sAcc, lsum);
  atomicAdd(&cAcc, (float)lcnt);
  __syncthreads();

  if (tid == 0) {
    float trip = (cAcc > 0.f) ? (sAcc / cAcc) : 0.f;
    float v = 0.f;
    for (int c = 0; c < C_; ++c) v += varsum[n * C_ + c] / counts[n * C_ + c];
    v *= (1.f / C_);
    float loss = v + trip + reg[n];          /* alpha=beta=gamma=1 */
    atomicAdd(out, loss * (1.f / (N_ * N_)));
  }
}

extern "C" void kernel_launch(void* const* d_in, const int* in_sizes, int n_in,
                              void* d_out, int out_size, void* d_ws, size_t ws_size,
                              hipStream_t stream) {
  const float* input_ = (const float*)d_in[0];
  const int*   target = (const int*)d_in[1];
  const int*   eattr  = (const int*)d_in[2];
  const int*   erep   = (const int*)d_in[3];
  float* out = (float*)d_out;
  float* ws  = (float*)d_ws;

  float* sums   = ws + WS_SUMS;
  float* counts = ws + WS_COUNTS;
  float* means  = ws + WS_MEANS;
  float* vsum   = ws + WS_VAR;
  float* reg    = ws + WS_REG;

  zero_k<<<1, 256, 0, stream>>>(ws, out);
  segsum_k<<<dim3(72, N_), 256, 0, stream>>>(input_, target, sums, counts);
  means_k<<<1, 128, 0, stream>>>(sums, counts, means, reg);
  var_k<<<dim3(144, N_), 256, 0, stream>>>(input_, target, means, vsum);
  final_k<<<N_, 256, 0, stream>>>(means, vsum, counts, reg, eattr, erep, out);
}